// CANLayer_15599321219072
// MI455X (gfx1250) — compile-verified
//
#include <hip/hip_runtime.h>
#include <hip/hip_bf16.h>

#define IN_C  64
#define OUT_C 64
#define EPS   (1.0f + 1e-06f)

typedef float v2f __attribute__((ext_vector_type(2)));
typedef float v8f __attribute__((ext_vector_type(8)));

// ---------------------------------------------------------------------------
// One 16-row block of (x @ W) via V_WMMA_F32_16X16X4_F32, K = 64 -> 16 steps
// per 16x16 output tile, 4 column tiles. HAS_ATT selects (at compile time)
// whether the two attention dot-products are accumulated per row.
//
// Fragment layouts per CDNA5 ISA (CDNA5 ISA):
//   A 16x4 f32 : lanes 0-15 hold row M=lane, K=0/1 in VGPR0/1;
//                lanes 16-31 hold row M=lane-16, K=2/3 in VGPR0/1.
//   B 4x16 f32 : lanes 0-15 hold col N=lane with K=0 (VGPR0), K=1 (VGPR1);
//                lanes 16-31 hold col N=lane-16 with K=2 (VGPR0), K=3 (VGPR1).
//   C/D 16x16  : VGPR r, lanes 0-15 -> (M=r,     N=lane),
//                        lanes 16-31 -> (M=r+8,  N=lane-16).
// ---------------------------------------------------------------------------
template <bool HAS_ATT>
__device__ __forceinline__ void wmma_block(
    const v2f a_frag[16],
    const float* __restrict__ w,      // [64,64] row-major
    const float* __restrict__ att,    // [128] (ignored unless HAS_ATT)
    float* __restrict__ store,        // [*,64] row-major destination
    float* __restrict__ s_src,
    float* __restrict__ s_tgt,
    float scale, int row0, int half, int lrow)
{
    // Preload this lane's attention scalars for all 4 column tiles.
    float att_s[4], att_t[4];
    if (HAS_ATT) {
#pragma unroll
        for (int nt = 0; nt < 4; ++nt) {
            att_s[nt] = att[nt * 16 + lrow];
            att_t[nt] = att[OUT_C + nt * 16 + lrow];
        }
    }

    float ssrc_acc[8];
    float stgt_acc[8];
#pragma unroll
    for (int r = 0; r < 8; ++r) { ssrc_acc[r] = 0.f; stgt_acc[r] = 0.f; }

#pragma unroll
    for (int nt = 0; nt < 4; ++nt) {
        const int col = nt * 16 + lrow;
        v8f c = {};
#pragma unroll
        for (int kk = 0; kk < 16; ++kk) {
            const int k0 = 4 * kk + 2 * half;
            v2f b;
            b.x = w[(size_t)k0 * OUT_C + col];
            b.y = w[(size_t)(k0 + 1) * OUT_C + col];
            c = __builtin_amdgcn_wmma_f32_16x16x4_f32(
                    /*neg_a=*/false, a_frag[kk],
                    /*neg_b=*/false, b,
                    /*c_mod=*/(short)0, c,
                    /*reuse_a=*/false, /*reuse_b=*/false);
        }
#pragma unroll
        for (int r = 0; r < 8; ++r) {
            const int row = row0 + r + half * 8;
            const float v = c[r];
            store[(size_t)row * OUT_C + col] = v * scale;
            if (HAS_ATT) {
                ssrc_acc[r] += v * att_s[nt];
                stgt_acc[r] += v * att_t[nt];
            }
        }
    }

    if (HAS_ATT) {
#pragma unroll
        for (int r = 0; r < 8; ++r) {
            float vs = ssrc_acc[r];
            float vt = stgt_acc[r];
            // reduce across the 16 lanes sharing this output row
#pragma unroll
            for (int m = 8; m >= 1; m >>= 1) {
                vs += __shfl_xor(vs, m, 32);
                vt += __shfl_xor(vt, m, 32);
            }
            if (lrow == 0) {
                const int row = row0 + r + half * 8;
                s_src[row] = vs;
                s_tgt[row] = vt;
            }
        }
    }
}

// ---------------------------------------------------------------------------
// Kernel A: fused node-level GEMMs. One wave handles 16 rows of x and
// produces xm_lower, xm_upper (to workspace), the four attention score
// vectors, and seeds out with (x @ w_lin) * EPS.
// ---------------------------------------------------------------------------
__global__ void __launch_bounds__(128)
node_gemm_kernel(const float* __restrict__ x,
                 const float* __restrict__ w_lower,
                 const float* __restrict__ a_lower,
                 const float* __restrict__ w_upper,
                 const float* __restrict__ a_upper,
                 const float* __restrict__ w_lin,
                 float* __restrict__ xm_l, float* __restrict__ xm_u,
                 float* __restrict__ sl_s, float* __restrict__ sl_t,
                 float* __restrict__ su_s, float* __restrict__ su_t,
                 float* __restrict__ out, int n_rows)
{
    const int wid  = (int)((blockIdx.x * blockDim.x + threadIdx.x) >> 5);
    const int lane = (int)(threadIdx.x & 31);
    const int row0 = wid * 16;
    if (row0 >= n_rows) return;               // wave-uniform: EXEC stays all-1

    const int half = lane >> 4;               // 0: K={0,1}, 1: K={2,3}
    const int lrow = lane & 15;               // M (for A) / N (for B,C)

    // Load all 16 A-fragments once; reused for all 12 output tiles.
    v2f a_frag[16];
    const float* xr = x + (size_t)(row0 + lrow) * IN_C + 2 * half;
#pragma unroll
    for (int kk = 0; kk < 16; ++kk)
        a_frag[kk] = *(const v2f*)(xr + 4 * kk);

    wmma_block<true >(a_frag, w_lower, a_lower, xm_l, sl_s, sl_t, 1.0f, row0, half, lrow);
    wmma_block<true >(a_frag, w_upper, a_upper, xm_u, su_s, su_t, 1.0f, row0, half, lrow);
    wmma_block<false>(a_frag, w_lin,   nullptr, out,  nullptr, nullptr, EPS, row0, half, lrow);
}

// ---------------------------------------------------------------------------
// Kernel B: edge scatter. One wave per edge. The edge id is forced wave-
// uniform via readfirstlane so the index/value/score loads lower to scalar
// (s_load) and the alpha computation is done once per wave; each lane then
// streams 2 channels of xm[j] (one global_load_b64) and issues 2 f32 atomics
// into out[i]. out and xm stay L2-resident (12.8 MB each << 192 MB L2), so
// this phase is L2-atomic bound — the minimum possible per-edge traffic.
// ---------------------------------------------------------------------------
__global__ void __launch_bounds__(256)
edge_scatter_kernel(const int* __restrict__ idx,     // [2,E]: row0 = i (tgt), row1 = j (src)
                    const float* __restrict__ vals,  // [E]
                    const float* __restrict__ xm,    // [N,64]
                    const float* __restrict__ s_src, // [N]
                    const float* __restrict__ s_tgt, // [N]
                    float* __restrict__ out,         // [N,64] accumulator
                    int E)
{
    const int e = __builtin_amdgcn_readfirstlane(
        (int)((blockIdx.x * blockDim.x + threadIdx.x) >> 5));
    const int lane = (int)(threadIdx.x & 31);
    if (e >= E) return;

    const int i = idx[e];                 // target
    const int j = idx[(size_t)E + e];     // source

    const float z = s_src[j] + s_tgt[i];
    const float elu = (z > 0.f) ? z : expm1f(z);
    const float alpha = elu * vals[e];

    const float2 m = ((const float2*)(xm + (size_t)j * OUT_C))[lane];
    float* o = out + (size_t)i * OUT_C + lane * 2;
    atomicAdd(o,     alpha * m.x);
    atomicAdd(o + 1, alpha * m.y);
}

// ---------------------------------------------------------------------------
// Kernel C: final ReLU over the accumulator.
// ---------------------------------------------------------------------------
__global__ void __launch_bounds__(256)
relu_kernel(float* __restrict__ out, size_t n)
{
    const size_t t = (size_t)blockIdx.x * blockDim.x + threadIdx.x;
    if (t < n) out[t] = fmaxf(out[t], 0.0f);
}

extern "C" void kernel_launch(void* const* d_in, const int* in_sizes, int n_in,
                              void* d_out, int out_size, void* d_ws, size_t ws_size,
                              hipStream_t stream) {
    const float* x        = (const float*)d_in[0];
    const int*   low_idx  = (const int*)  d_in[1];
    const float* low_val  = (const float*)d_in[2];
    const int*   up_idx   = (const int*)  d_in[3];
    const float* up_val   = (const float*)d_in[4];
    const float* w_lower  = (const float*)d_in[5];
    const float* a_lower  = (const float*)d_in[6];
    const float* w_upper  = (const float*)d_in[7];
    const float* a_upper  = (const float*)d_in[8];
    const float* w_lin    = (const float*)d_in[9];
    float*       out      = (float*)d_out;

    const int N = in_sizes[0] / IN_C;   // 50000
    const int E = in_sizes[2];          // 800000

    // Workspace layout (floats): xm_l[N*64] | xm_u[N*64] | 4 x s[N]
    float* ws   = (float*)d_ws;
    float* xm_l = ws;
    float* xm_u = xm_l + (size_t)N * OUT_C;
    float* sl_s = xm_u + (size_t)N * OUT_C;
    float* sl_t = sl_s + N;
    float* su_s = sl_t + N;
    float* su_t = su_s + N;

    // Kernel A: 1 wave per 16 rows, 4 waves (128 thr) per block.
    {
        const int waves  = (N + 15) / 16;
        const int blocks = (waves + 3) / 4;
        node_gemm_kernel<<<blocks, 128, 0, stream>>>(
            x, w_lower, a_lower, w_upper, a_upper, w_lin,
            xm_l, xm_u, sl_s, sl_t, su_s, su_t, out, N);
    }

    // Kernel B: 1 wave per edge, 8 waves (256 thr) per block.
    {
        const int blocks = (E + 7) / 8;
        edge_scatter_kernel<<<blocks, 256, 0, stream>>>(
            low_idx, low_val, xm_l, sl_s, sl_t, out, E);
        edge_scatter_kernel<<<blocks, 256, 0, stream>>>(
            up_idx, up_val, xm_u, su_s, su_t, out, E);
    }

    // Kernel C: ReLU.
    {
        const size_t n = (size_t)N * OUT_C;
        const int blocks = (int)((n + 255) / 256);
        relu_kernel<<<blocks, 256, 0, stream>>>(out, n);
    }
}